// GRU_GAT_Model_18253611008142
// MI455X (gfx1250) — compile-verified
//
#include <hip/hip_runtime.h>
#include <math.h>
#include <stdint.h>

// ---------------------------------------------------------------------------
// GRU+GAT model for MI455X (gfx1250, wave32).
// Dense GEMMs (GRU gates, MLP1, LIN2) use V_WMMA_F32_16X16X4_F32: fp32 matrix
// pipe, batch=16 maps exactly onto the WMMA N dimension. LIN2 streams 134 MB
// of fp32 weights (HBM-bound); its activation panel (64 KB) is staged into
// LDS once per workgroup via GLOBAL_LOAD_ASYNC_TO_LDS_B128 (ASYNCcnt path),
// so the K-loop reads B-operands from LDS and global traffic is weights only.
// GAT uses atomics-based segment-softmax (scatter max / scatter add), one
// wave per (batch, edge) with lane==feature for coalesced accumulation.
// ---------------------------------------------------------------------------

typedef __attribute__((ext_vector_type(2))) float v2f;
typedef __attribute__((ext_vector_type(8))) float v8f;

constexpr int kB   = 16;
constexpr int kN   = 4096;
constexpr int kE   = 65536;
constexpr int kIN  = 256;
constexpr int kH   = 512;
constexpr int kG   = 32;     // G1 == G2 == 32
constexpr int kM1  = 1024;
constexpr int kM2  = 32768;
constexpr int kGOD = 8;
constexpr int kET  = kE + kN;   // edges + self loops

__device__ __forceinline__ float lrelu02(float x) { return x >= 0.f ? x : 0.2f * x; }
__device__ __forceinline__ float sigm(float x)    { return 1.f / (1.f + __expf(-x)); }

__device__ void atomicMaxF(float* addr, float val) {
  unsigned* ua = reinterpret_cast<unsigned*>(addr);
  unsigned old = *ua;
  while (__uint_as_float(old) < val) {
    unsigned assumed = old;
    old = atomicCAS(ua, assumed, __float_as_uint(val));
    if (old == assumed) break;
  }
}

// Wave-level fp32 WMMA tile: D[16 m][16 n] = sum_k W[m0+m][k] * XT[k][n] + C.
// A layout: lane L -> row M = L&15, K = 2*(L>=16)+{0,1} in the 2 A VGPRs.
// B layout (mirrored): lane L -> col N = L&15, same K split.
// C/D layout: VGPR r -> (M = r + 8*(L>=16), N = L&15).
__device__ __forceinline__ v8f wmma_tile_f32(const float* __restrict__ W, int ldw,
                                             const float* __restrict__ XT, int K,
                                             v8f acc, int lane) {
  const int m  = lane & 15;
  const int kh = (lane >> 4) << 1;   // 0 or 2
  for (int k = 0; k < K; k += 4) {
    v2f a, b;
    const float* wp = W + (size_t)m * ldw + k + kh;
    a.x = wp[0];  a.y = wp[1];
    const float* xp = XT + (k + kh) * 16 + m;
    b.x = xp[0];  b.y = xp[16];
    acc = __builtin_amdgcn_wmma_f32_16x16x4_f32(false, a, false, b,
                                                (short)0, acc, false, false);
  }
  return acc;
}

// ---------------------------------------------------------------------------
// Small transpose: x [16,256] -> xT [256,16]; hidden [16,512] -> hT [512,16].
__global__ void k_transpose(const float* __restrict__ x, const float* __restrict__ hid,
                            float* __restrict__ xT, float* __restrict__ hT) {
  int idx = blockIdx.x * blockDim.x + threadIdx.x;
  if (idx < kB * kIN) {
    int b = idx / kIN, k = idx % kIN;
    xT[k * 16 + b] = x[idx];
  } else if (idx < kB * kIN + kB * kH) {
    int t = idx - kB * kIN;
    int b = t / kH, k = t % kH;
    hT[k * 16 + b] = hid[t];
  }
}

// ---------------------------------------------------------------------------
// GAT node pass: h = v @ W^T, per-node attention sums, init reductions.
__global__ void k_gat_node(const float* __restrict__ vin, int fin,
                           const float* __restrict__ W,
                           const float* __restrict__ a_src, const float* __restrict__ a_dst,
                           float* __restrict__ h, float* __restrict__ ssrc,
                           float* __restrict__ sdst, float* __restrict__ m,
                           float* __restrict__ s, float* __restrict__ acc) {
  int idx = blockIdx.x * blockDim.x + threadIdx.x;   // over B*N
  if (idx >= kB * kN) return;
  const float* v = vin + (size_t)idx * fin;
  float ss = 0.f, sd = 0.f;
  for (int j = 0; j < kG; ++j) {
    float hj = 0.f;
    const float* wr = W + j * fin;
    for (int i = 0; i < fin; ++i) hj += v[i] * wr[i];
    h[(size_t)idx * kG + j]   = hj;
    acc[(size_t)idx * kG + j] = 0.f;
    ss += hj * a_src[j];
    sd += hj * a_dst[j];
  }
  ssrc[idx] = ss;
  sdst[idx] = sd;
  m[idx] = -INFINITY;
  s[idx] = 0.f;
}

// Edge pass 1: segment max of leaky_relu attention logits.
__global__ void k_gat_edge_max(const int* __restrict__ ei,
                               const float* __restrict__ ssrc, const float* __restrict__ sdst,
                               float* __restrict__ m) {
  int idx = blockIdx.x * blockDim.x + threadIdx.x;   // over B*(E+N)
  if (idx >= kB * kET) return;
  int b = idx / kET, e = idx % kET;
  int src = (e < kE) ? ei[e]      : (e - kE);
  int dst = (e < kE) ? ei[kE + e] : (e - kE);
  float ee = lrelu02(ssrc[b * kN + src] + sdst[b * kN + dst]);
  atomicMaxF(&m[b * kN + dst], ee);
}

// Edge pass 2: one wave per (batch, edge); lane == feature. Coalesced
// atomic accumulation of w * h[src] into acc[dst]; lane 0 adds normalizer.
__global__ void k_gat_edge_acc(const int* __restrict__ ei,
                               const float* __restrict__ ssrc, const float* __restrict__ sdst,
                               const float* __restrict__ m, const float* __restrict__ h,
                               float* __restrict__ s, float* __restrict__ acc) {
  int wid  = blockIdx.x * (blockDim.x >> 5) + (threadIdx.x >> 5);
  int lane = threadIdx.x & 31;
  if (wid >= kB * kET) return;
  int b = wid / kET, e = wid % kET;
  int src = (e < kE) ? ei[e]      : (e - kE);
  int dst = (e < kE) ? ei[kE + e] : (e - kE);
  int ns = b * kN + src, nd = b * kN + dst;
  float ee = lrelu02(ssrc[ns] + sdst[nd]);
  float w  = __expf(ee - m[nd]);
  atomicAdd(&acc[(size_t)nd * kG + lane], w * h[(size_t)ns * kG + lane]);
  if (lane == 0) atomicAdd(&s[nd], w);
}

// Node finalize: out = acc / s + bias.
__global__ void k_gat_node_fin(const float* __restrict__ acc, const float* __restrict__ s,
                               const float* __restrict__ bias, float* __restrict__ out) {
  int idx = blockIdx.x * blockDim.x + threadIdx.x;   // over B*N*G
  if (idx >= kB * kN * kG) return;
  int f = idx & (kG - 1);
  int node = idx / kG;
  out[idx] = acc[idx] / s[node] + bias[f];
}

// ---------------------------------------------------------------------------
// GRUCell: one wave per 16 hidden units; six 16x16 WMMA-accumulated tiles.
__global__ void __launch_bounds__(32)
k_gru(const float* __restrict__ w_ih, const float* __restrict__ w_hh,
      const float* __restrict__ b_ih, const float* __restrict__ b_hh,
      const float* __restrict__ xT,  const float* __restrict__ hT,
      const float* __restrict__ hidden,
      float* __restrict__ nh_out, float* __restrict__ nhT) {
  const int lane = threadIdx.x;
  const int j0   = blockIdx.x * 16;
  v8f ir = {}, iz = {}, in_ = {}, hr = {}, hz = {}, hn = {};
  ir  = wmma_tile_f32(w_ih + (size_t)(0 * kH + j0) * kIN, kIN, xT, kIN, ir,  lane);
  iz  = wmma_tile_f32(w_ih + (size_t)(1 * kH + j0) * kIN, kIN, xT, kIN, iz,  lane);
  in_ = wmma_tile_f32(w_ih + (size_t)(2 * kH + j0) * kIN, kIN, xT, kIN, in_, lane);
  hr  = wmma_tile_f32(w_hh + (size_t)(0 * kH + j0) * kH,  kH,  hT, kH,  hr,  lane);
  hz  = wmma_tile_f32(w_hh + (size_t)(1 * kH + j0) * kH,  kH,  hT, kH,  hz,  lane);
  hn  = wmma_tile_f32(w_hh + (size_t)(2 * kH + j0) * kH,  kH,  hT, kH,  hn,  lane);
  const int n  = lane & 15;
  const int mo = (lane >> 4) << 3;
  for (int r = 0; r < 8; ++r) {
    int j = j0 + r + mo;
    float rg = sigm(ir[r] + b_ih[j]          + hr[r] + b_hh[j]);
    float zg = sigm(iz[r] + b_ih[kH + j]     + hz[r] + b_hh[kH + j]);
    float ng = tanhf(in_[r] + b_ih[2*kH + j] + rg * (hn[r] + b_hh[2*kH + j]));
    float hp = hidden[n * kH + j];
    float nv = (1.f - zg) * ng + zg * hp;
    nh_out[n * kH + j] = nv;     // next_hidden output (row-major [B,H])
    nhT[j * 16 + n]    = nv;     // transposed for MLP1 B-operand
  }
}

// MLP layer 1 (Linear + PReLU + eval BN), one wave per 16 outputs.
__global__ void __launch_bounds__(32)
k_mlp1(const float* __restrict__ w, const float* __restrict__ bias,
       const float* __restrict__ prelu, const float* __restrict__ gg,
       const float* __restrict__ be, const float* __restrict__ nhT,
       float* __restrict__ h1T) {
  const int lane = threadIdx.x;
  const int j0   = blockIdx.x * 16;
  v8f acc = {};
  acc = wmma_tile_f32(w + (size_t)j0 * kH, kH, nhT, kH, acc, lane);
  const float bninv = rsqrtf(1.f + 1e-5f);
  const int n = lane & 15, mo = (lane >> 4) << 3;
  for (int r = 0; r < 8; ++r) {
    int j = j0 + r + mo;
    float t = acc[r] + bias[j];
    t = t >= 0.f ? t : prelu[j] * t;
    t = t * bninv * gg[j] + be[j];
    h1T[j * 16 + n] = t;         // transposed for LIN2 B-operand
  }
}

// MLP layer 2: the HBM-bound GEMM (134 MB fp32 weights). 8 waves/block,
// one wave per 16 output neurons. The activation panel h1T [1024 x 16]
// (64 KB) is staged into LDS once per block with async-to-LDS copies
// (ASYNCcnt), then the K-loop reads B-operands from LDS while weights
// stream from HBM with prefetch.
__global__ void __launch_bounds__(256)
k_lin2(const float* __restrict__ w, const float* __restrict__ bias,
       const float* __restrict__ prelu, const float* __restrict__ gg,
       const float* __restrict__ be, const float* __restrict__ h1T,
       float* __restrict__ o2) {
  __shared__ float tile[kM1 * 16];   // 64 KB activation panel, [k][n]

  // Cooperative async stage: 16384 floats = 4096 x b128, 16 per thread.
  for (int i = threadIdx.x; i < (kM1 * 16) / 4; i += 256) {
    unsigned lds_off = (unsigned)(uintptr_t)(&tile[i * 4]);
    const float* gp = h1T + (size_t)i * 4;
    asm volatile("global_load_async_to_lds_b128 %0, %1, off"
                 :: "v"(lds_off), "v"(gp) : "memory");
  }
  asm volatile("s_wait_asynccnt 0x0" ::: "memory");
  __syncthreads();

  const int lane = threadIdx.x & 31;
  const int tileid = blockIdx.x * (blockDim.x >> 5) + (threadIdx.x >> 5);
  const int m0   = tileid * 16;
  const int mr   = lane & 15;
  const int kh   = (lane >> 4) << 1;
  v8f acc = {};
  const float* wrow = w + (size_t)(m0 + mr) * kM1 + kh;
  const float* lp   = &tile[kh * 16 + mr];
  for (int k = 0; k < kM1; k += 4) {
    v2f a, b;
    a.x = wrow[k];  a.y = wrow[k + 1];
    __builtin_prefetch(wrow + k + 512, 0, 1);   // global_prefetch_b8 ahead in K
    b.x = lp[k * 16];  b.y = lp[k * 16 + 16];   // ds_load from staged panel
    acc = __builtin_amdgcn_wmma_f32_16x16x4_f32(false, a, false, b,
                                                (short)0, acc, false, false);
  }
  const float bninv = rsqrtf(1.f + 1e-5f);
  const int n = lane & 15, mo = (lane >> 4) << 3;
  for (int r = 0; r < 8; ++r) {
    int j = m0 + r + mo;
    float t = acc[r] + bias[j];
    t = t >= 0.f ? t : prelu[j] * t;
    t = t * bninv * gg[j] + be[j];
    o2[(size_t)n * kM2 + j] = t;   // [B, M2] row-major
  }
}

// Output head: y[b,n,:] = [gru_out(8) | gat1(32)] @ w_out^T + b_out.
__global__ void k_final(const float* __restrict__ o2, const float* __restrict__ gat1,
                        const float* __restrict__ w_out, const float* __restrict__ b_out,
                        float* __restrict__ y) {
  int idx = blockIdx.x * blockDim.x + threadIdx.x;   // over B*N
  if (idx >= kB * kN) return;
  int b = idx / kN, n = idx % kN;
  const float* go = o2 + (size_t)b * kM2 + n * kGOD;
  const float* ga = gat1 + (size_t)idx * kG;
  for (int c = 0; c < 3; ++c) {
    const float* wr = w_out + c * (kGOD + kG);
    float sm = b_out[c];
    for (int j = 0; j < kGOD; ++j) sm += go[j] * wr[j];
    for (int f = 0; f < kG;   ++f) sm += ga[f] * wr[kGOD + f];
    y[(size_t)b * (kN * 3) + n * 3 + c] = sm;
  }
}

// ---------------------------------------------------------------------------
extern "C" void kernel_launch(void* const* d_in, const int* in_sizes, int n_in,
                              void* d_out, int out_size, void* d_ws, size_t ws_size,
                              hipStream_t stream) {
  (void)in_sizes; (void)n_in; (void)out_size; (void)ws_size;

  const float* x    = (const float*)d_in[0];
  const float* sv   = (const float*)d_in[1];
  const float* hid  = (const float*)d_in[2];
  const int*   ei   = (const int*)  d_in[3];
  const float* W0   = (const float*)d_in[4];
  const float* a0s  = (const float*)d_in[5];
  const float* a0d  = (const float*)d_in[6];
  const float* b0   = (const float*)d_in[7];
  const float* W1   = (const float*)d_in[8];
  const float* a1s  = (const float*)d_in[9];
  const float* a1d  = (const float*)d_in[10];
  const float* b1   = (const float*)d_in[11];
  const float* w_ih = (const float*)d_in[12];
  const float* w_hh = (const float*)d_in[13];
  const float* b_ih = (const float*)d_in[14];
  const float* b_hh = (const float*)d_in[15];
  const float* l1w  = (const float*)d_in[16];
  const float* l1b  = (const float*)d_in[17];
  const float* p1   = (const float*)d_in[18];
  const float* g1   = (const float*)d_in[19];
  const float* be1  = (const float*)d_in[20];
  const float* l2w  = (const float*)d_in[21];
  const float* l2b  = (const float*)d_in[22];
  const float* p2   = (const float*)d_in[23];
  const float* g2   = (const float*)d_in[24];
  const float* be2  = (const float*)d_in[25];
  const float* wout = (const float*)d_in[26];
  const float* bout = (const float*)d_in[27];

  float* ws = (float*)d_ws;
  size_t off = 0;
  float* h_buf = ws + off; off += (size_t)kB * kN * kG;
  float* acc   = ws + off; off += (size_t)kB * kN * kG;
  float* ssrc  = ws + off; off += (size_t)kB * kN;
  float* sdst  = ws + off; off += (size_t)kB * kN;
  float* mbuf  = ws + off; off += (size_t)kB * kN;
  float* sbuf  = ws + off; off += (size_t)kB * kN;
  float* gat0  = ws + off; off += (size_t)kB * kN * kG;
  float* gat1  = ws + off; off += (size_t)kB * kN * kG;
  float* xT    = ws + off; off += (size_t)kIN * 16;
  float* hT    = ws + off; off += (size_t)kH * 16;
  float* nhT   = ws + off; off += (size_t)kH * 16;
  float* h1T   = ws + off; off += (size_t)kM1 * 16;
  float* o2    = ws + off; off += (size_t)kB * kM2;

  float* y  = (float*)d_out;            // [B, N*3]
  float* nh = y + (size_t)kB * kN * 3;  // [B, H]

  k_transpose<<<(kB * (kIN + kH) + 255) / 256, 256, 0, stream>>>(x, hid, xT, hT);

  // GAT layer 0 (F_in = 3)
  k_gat_node<<<(kB * kN + 255) / 256, 256, 0, stream>>>(sv, 3, W0, a0s, a0d,
                                                        h_buf, ssrc, sdst, mbuf, sbuf, acc);
  k_gat_edge_max<<<(kB * kET + 255) / 256, 256, 0, stream>>>(ei, ssrc, sdst, mbuf);
  k_gat_edge_acc<<<(kB * kET + 7) / 8, 256, 0, stream>>>(ei, ssrc, sdst, mbuf, h_buf, sbuf, acc);
  k_gat_node_fin<<<(kB * kN * kG + 255) / 256, 256, 0, stream>>>(acc, sbuf, b0, gat0);

  // GAT layer 1 (F_in = 32)
  k_gat_node<<<(kB * kN + 255) / 256, 256, 0, stream>>>(gat0, kG, W1, a1s, a1d,
                                                        h_buf, ssrc, sdst, mbuf, sbuf, acc);
  k_gat_edge_max<<<(kB * kET + 255) / 256, 256, 0, stream>>>(ei, ssrc, sdst, mbuf);
  k_gat_edge_acc<<<(kB * kET + 7) / 8, 256, 0, stream>>>(ei, ssrc, sdst, mbuf, h_buf, sbuf, acc);
  k_gat_node_fin<<<(kB * kN * kG + 255) / 256, 256, 0, stream>>>(acc, sbuf, b1, gat1);

  // Dense path: GRU -> MLP1 -> LIN2 (all fp32 WMMA)
  k_gru<<<kH / 16, 32, 0, stream>>>(w_ih, w_hh, b_ih, b_hh, xT, hT, hid, nh, nhT);
  k_mlp1<<<kM1 / 16, 32, 0, stream>>>(l1w, l1b, p1, g1, be1, nhT, h1T);
  k_lin2<<<kM2 / 16 / 8, 256, 0, stream>>>(l2w, l2b, p2, g2, be2, h1T, o2);

  k_final<<<(kB * kN + 255) / 256, 256, 0, stream>>>(o2, gat1, wout, bout, y);
}